// EventFilter_87857851007244
// MI455X (gfx1250) — compile-verified
//
#include <hip/hip_runtime.h>

#define D_ 32
#define H_ 32
#define W_ 32
#define V_ (D_*H_*W_)
#define KCAP 512
#define MAXEV 100

typedef float v2f __attribute__((ext_vector_type(2)));
typedef float v8f __attribute__((ext_vector_type(8)));

__global__ __launch_bounds__(256) void event_filter_kernel(const float* __restrict__ x,
                                                           float* __restrict__ out) {
    __shared__ float    s_val[KCAP];
    __shared__ int      s_idx[KCAP];
    __shared__ float    s_cz[KCAP], s_cy[KCAP], s_cx[KCAP], s_n2[KCAP];
    __shared__ unsigned s_adj[KCAP * 16];      // 512 rows x 512 bits
    __shared__ unsigned s_vox[V_ / 32];        // voxel keep-mask
    __shared__ unsigned s_keep[16];
    __shared__ int      s_cnt;
    __shared__ int      s_nz;

    const int tid   = threadIdx.x;
    const int frame = blockIdx.x;
    const float* eIn = x + (size_t)frame * (2 * V_);
    const float* mIn = eIn + V_;
    float* eOut = out + (size_t)frame * (2 * V_);
    float* mOut = eOut + V_;

    // ---- phase 0: init LDS ----
    for (int i = tid; i < KCAP * 16; i += 256) s_adj[i] = 0u;
    for (int i = tid; i < V_ / 32; i += 256)   s_vox[i] = 0u;
    if (tid == 0) { s_cnt = 0; s_nz = 0; }
    __syncthreads();

    // ---- phase 1: collect positive-energy voxels ----
    for (int v = tid; v < V_; v += 256) {
        float e = eIn[v];
        if ((tid & 31) == 0) __builtin_prefetch(mIn + v, 0, 1);  // stage ch1 into L2
        if (e != 0.0f) atomicOr(&s_nz, 1);
        if (e > 0.0f) {
            int p = atomicAdd(&s_cnt, 1);
            if (p < KCAP) { s_val[p] = e; s_idx[p] = v; }
        }
    }
    __syncthreads();
    int n = s_cnt; if (n > KCAP) n = KCAP;
    for (int s = tid; s < KCAP; s += 256)
        if (s >= n) { s_val[s] = -1.0f; s_idx[s] = 0; }
    __syncthreads();

    // ---- phase 2: bitonic sort (value desc, index asc) ----
    for (int k = 2; k <= KCAP; k <<= 1) {
        for (int j = k >> 1; j > 0; j >>= 1) {
            for (int e0 = tid; e0 < KCAP; e0 += 256) {
                int p = e0 ^ j;
                if (p > e0) {
                    float va = s_val[e0], vb = s_val[p];
                    int   ia = s_idx[e0], ib = s_idx[p];
                    bool before  = (va > vb) || (va == vb && ia < ib);
                    bool forward = ((e0 & k) == 0);
                    if (forward ? !before : before) {
                        s_val[e0] = vb; s_val[p] = va;
                        s_idx[e0] = ib; s_idx[p] = ia;
                    }
                }
            }
            __syncthreads();
        }
    }

    // ---- phase 2.5: coordinates + squared norms ----
    for (int s = tid; s < KCAP; s += 256) {
        int id = (s < n) ? s_idx[s] : 0;
        float z  = (float)(id >> 10);
        float y  = (float)((id >> 5) & 31);
        float xx = (float)(id & 31);
        s_cz[s] = z; s_cy[s] = y; s_cx[s] = xx;
        s_n2[s] = z * z + y * y + xx * xx;
    }
    __syncthreads();

    // ---- phase 3: adjacency via V_WMMA_F32_16X16X4_F32 Gram tiles ----
    const int lane = tid & 31;
    const int wave = tid >> 5;
    const int half = lane >> 4;   // 0: lanes 0-15 (K=0,1) ; 1: lanes 16-31 (K=2,3)
    const int ln   = lane & 15;
    int nTiles = (n + 15) >> 4;
    int nPairs = nTiles * nTiles;
    for (int tp = wave; tp < nPairs; tp += 8) {
        int ti = tp / nTiles;
        int tj = tp - ti * nTiles;
        int gi0 = ti * 16, gj0 = tj * 16;
        v2f a, b;
        a.x = half ? s_cx[gi0 + ln] : s_cz[gi0 + ln];
        a.y = half ? 0.0f           : s_cy[gi0 + ln];
        b.x = half ? s_cx[gj0 + ln] : s_cz[gj0 + ln];
        b.y = half ? 0.0f           : s_cy[gj0 + ln];
        v8f c = {0.f, 0.f, 0.f, 0.f, 0.f, 0.f, 0.f, 0.f};
        v8f d = __builtin_amdgcn_wmma_f32_16x16x4_f32(
                    false, a, false, b, (short)0, c, false, false);
        int gj = gj0 + ln;
        float n2j = s_n2[gj];
#pragma unroll
        for (int r = 0; r < 8; ++r) {
            int gi = gi0 + r + half * 8;
            float dist2 = s_n2[gi] + n2j - 2.0f * d[r];
            bool sup = (gj > gi) && (dist2 < 4.0f);
            unsigned bal = __builtin_amdgcn_ballot_w32(sup);
            if (ln == 0) {
                unsigned bits = half ? (bal >> 16) : (bal & 0xFFFFu);
                if (bits) atomicOr(&s_adj[gi * 16 + (tj >> 1)], bits << ((tj & 1) * 16));
            }
        }
    }
    __syncthreads();

    // ---- phase 4: serial greedy NMS as bitmask sweep (wave 0) ----
    if (wave == 0) {
        unsigned kw = 0;
        if (lane < 16) {
            int rem = n - lane * 32;
            kw = rem <= 0 ? 0u : (rem >= 32 ? 0xFFFFFFFFu : ((1u << rem) - 1u));
        }
        for (int i = 0; i < n; ++i) {
            unsigned srcw = __shfl(kw, i >> 5, 32);
            if (((srcw >> (i & 31)) & 1u) && lane < 16)
                kw &= ~s_adj[i * 16 + lane];
        }
        unsigned pc = (lane < 16) ? (unsigned)__popc(kw) : 0u;
        pc += __shfl_down(pc, 16, 32);
        pc += __shfl_down(pc, 8, 32);
        pc += __shfl_down(pc, 4, 32);
        pc += __shfl_down(pc, 2, 32);
        pc += __shfl_down(pc, 1, 32);
        unsigned total = __shfl(pc, 0, 32);
        if (total > MAXEV && lane < 16) {   // keep only sorted positions < 100
            int rem = MAXEV - lane * 32;
            unsigned m = rem <= 0 ? 0u : (rem >= 32 ? 0xFFFFFFFFu : ((1u << rem) - 1u));
            kw &= m;
        }
        if (lane < 16) s_keep[lane] = kw;
    }
    __syncthreads();

    // ---- phase 5: scatter kept voxels into bitmask ----
    for (int s = tid; s < n; s += 256) {
        if ((s_keep[s >> 5] >> (s & 31)) & 1u) {
            int id = s_idx[s];
            atomicOr(&s_vox[id >> 5], 1u << (id & 31));
        }
    }
    __syncthreads();

    // ---- phase 6: masked streaming write (float4) ----
    bool pass = (s_nz == 0);   // empty frame passes through unchanged
    const float4* e4  = (const float4*)eIn;
    const float4* m4  = (const float4*)mIn;
    float4*       eo4 = (float4*)eOut;
    float4*       mo4 = (float4*)mOut;
    for (int q = tid; q < V_ / 4; q += 256) {
        float4 ev = e4[q];
        float4 mv = m4[q];
        int vb = q * 4;
        unsigned wb = s_vox[vb >> 5] >> (vb & 31);
        if (!pass) {
            if (!(wb & 1u)) { ev.x = 0.f; mv.x = 0.f; }
            if (!(wb & 2u)) { ev.y = 0.f; mv.y = 0.f; }
            if (!(wb & 4u)) { ev.z = 0.f; mv.z = 0.f; }
            if (!(wb & 8u)) { ev.w = 0.f; mv.w = 0.f; }
        }
        eo4[q] = ev;
        mo4[q] = mv;
    }
}

extern "C" void kernel_launch(void* const* d_in, const int* in_sizes, int n_in,
                              void* d_out, int out_size, void* d_ws, size_t ws_size,
                              hipStream_t stream) {
    const float* x = (const float*)d_in[0];
    float* out = (float*)d_out;
    int frames = in_sizes[0] / (2 * V_);   // B*T
    hipLaunchKernelGGL(event_filter_kernel, dim3(frames), dim3(256), 0, stream, x, out);
}